// NeuralSplineBase_51651276701818
// MI455X (gfx1250) — compile-verified
//
#include <hip/hip_runtime.h>

typedef __attribute__((ext_vector_type(2))) float v2f;
typedef __attribute__((ext_vector_type(4))) float v4f;
typedef __attribute__((ext_vector_type(8))) float v8f;

#define NK    10          // knots
#define NSEG  9           // segments
#define NCH   36          // 4*3*3 spline channels
#define HSTEP (1.0f/9.0f) // knot spacing
#define HW4   262144      // (1024*1024)/4 float4s per plane

// ---------------------------------------------------------------------------
// Kernel 1: one workgroup. Builds the constant spline operator (8x8 tridiag
// inverse via Gauss-Jordan), then computes M = mat10 @ ys^T as an fp32 GEMM on
// the matrix cores (V_WMMA_F32_16X16X4_F32, one wave per 16-col tile, 3
// K-steps), then emits per-segment cubic coefficients + the 36x255 spline LUT.
// ---------------------------------------------------------------------------
__global__ __launch_bounds__(128)
void spline_precompute(const float* __restrict__ ys,
                       v4f* __restrict__ coef,          // [NCH][NSEG] {a,b,c,d}
                       float* __restrict__ splines_out) // [3][4][3][255]
{
    __shared__ float sInv[8][8];
    __shared__ float sMat[16][16];   // padded mat10 (A operand)
    __shared__ float sYsT[16][48];   // padded ys^T: [k][ch] (B operand)
    __shared__ float sM[NCH][NK];    // second derivatives
    __shared__ float sYs[NCH][NK];   // ys + identity
    const int tid = threadIdx.x;

    // --- stage 0: inv of tridiag(1,4,1) 8x8 (constant, trivial cost) ---
    if (tid == 0) {
        float T[8][8];
        for (int i = 0; i < 8; ++i)
            for (int j = 0; j < 8; ++j) {
                T[i][j]    = (i == j) ? 4.f : ((i - j == 1 || j - i == 1) ? 1.f : 0.f);
                sInv[i][j] = (i == j) ? 1.f : 0.f;
            }
        for (int k = 0; k < 8; ++k) {
            float p = 1.0f / T[k][k];
            for (int j = 0; j < 8; ++j) { T[k][j] *= p; sInv[k][j] *= p; }
            for (int i = 0; i < 8; ++i) if (i != k) {
                float f = T[i][k];
                for (int j = 0; j < 8; ++j) { T[i][j] -= f * T[k][j]; sInv[i][j] -= f * sInv[k][j]; }
            }
        }
    }
    __syncthreads();

    // --- stage 0b: mat10 = [0; 6*inv(T)/h^2; 0] @ secondDiff, padded 16x16 ---
    for (int t = tid; t < 16 * 16; t += blockDim.x) {
        int m = t >> 4, n = t & 15;
        float v = 0.f;
        if (m >= 1 && m <= 8 && n < NK) {
            float s = 0.f;
            if (n <= 7)           s += sInv[m - 1][n];
            if (n >= 1 && n <= 8) s -= 2.f * sInv[m - 1][n - 1];
            if (n >= 2)           s += sInv[m - 1][n - 2];
            v = 486.f * s;        // 6/h^2 = 6*81
        }
        sMat[m][n] = v;
    }
    // --- stage 1: ys + identity ramp, transposed & zero-padded to 16x48 ---
    for (int t = tid; t < 16 * 48; t += blockDim.x) {
        int k = t / 48, ch = t % 48;
        float v = 0.f;
        if (k < NK && ch < NCH) {
            v = ys[ch * NK + k] + (float)k * HSTEP;
            sYs[ch][k] = v;
        }
        sYsT[k][ch] = v;
    }
    __syncthreads();

    // --- stage 2: M(10x36) = mat10(10x10) @ ysT(10x36) via WMMA f32 16x16x4 ---
    const int lane = tid & 31;
    const int wid  = tid >> 5;
    if (wid < 3) {                      // 3 column tiles, one wave each (EXEC all-1s)
        const int row   = lane & 15;
        const int khalf = (lane >> 4) * 2;   // lanes 0-15 -> K{0,1}; 16-31 -> K{2,3}
        const int colB  = wid * 16;
        v8f acc = {};
        #pragma unroll
        for (int ks = 0; ks < 3; ++ks) {     // K = 0..11 covers NK=10 (zero padded)
            int k0 = ks * 4 + khalf;
            v2f a, b;
            a.x = sMat[row][k0];
            a.y = sMat[row][k0 + 1];
            b.x = sYsT[k0][colB + row];
            b.y = sYsT[k0 + 1][colB + row];
            acc = __builtin_amdgcn_wmma_f32_16x16x4_f32(
                /*neg_a=*/false, a, /*neg_b=*/false, b,
                /*c_mod=*/(short)0, acc, /*reuse_a=*/false, /*reuse_b=*/false);
        }
        #pragma unroll
        for (int r = 0; r < 8; ++r) {
            int m  = r + ((lane >> 4) << 3); // lanes 16-31 hold rows 8..15
            int ch = colB + row;
            if (m < NK && ch < NCH) sM[ch][m] = acc[r];
        }
    }
    __syncthreads();

    // --- stage 3: cubic coefficients per (channel, segment) ---
    for (int t = tid; t < NCH * NSEG; t += blockDim.x) {
        int ch = t / NSEG, s = t % NSEG;
        float M0 = sM[ch][s], M1 = sM[ch][s + 1];
        float y0 = sYs[ch][s], y1 = sYs[ch][s + 1];
        v4f cf;
        cf.x = (M1 - M0) * 1.5f;                            // /(6h), h=1/9
        cf.y = M0 * 0.5f;
        cf.z = (y1 - y0) * 9.0f - (M1 + 2.f * M0) * (HSTEP / 6.f);
        cf.w = y0;
        coef[t] = cf;
    }

    // --- stage 4: splines LUT, output order [E=3][B=4][C=3][255] ---
    for (int t = tid; t < NCH * 255; t += blockDim.x) {
        int co = t / 255, p = t % 255;
        int e = co / 12, b = (co / 3) % 4, c = co % 3;
        int ch = (b * 3 + e) * 3 + c;        // coeff channel order [B][E][C]
        float x = (float)p * (1.0f / 255.0f);
        int s = (int)(x * 9.0f);
        s = (s < 0) ? 0 : ((s > 8) ? 8 : s);
        float xf = x - (float)s * HSTEP;
        float M0 = sM[ch][s], M1 = sM[ch][s + 1];
        float y0 = sYs[ch][s], y1 = sYs[ch][s + 1];
        float ca = (M1 - M0) * 1.5f;
        float cb = M0 * 0.5f;
        float cc = (y1 - y0) * 9.0f - (M1 + 2.f * M0) * (HSTEP / 6.f);
        splines_out[t] = ((ca * xf + cb) * xf + cc) * xf + y0;
    }
}

// ---------------------------------------------------------------------------
// Kernel 2: bandwidth-bound evaluation. Each block owns 1024 pixels of a
// single (b,c) plane; 27 coefficient float4s staged in LDS; 128-bit NT loads
// of batch, 3x 128-bit NT stores to [E][B][C][H][W].
// ---------------------------------------------------------------------------
__global__ __launch_bounds__(256)
void spline_eval(const v4f* __restrict__ batch,
                 const v4f* __restrict__ coef,
                 v4f* __restrict__ out)
{
    __shared__ v4f sC[3][NSEG];
    const int tid = threadIdx.x;
    const int gid = blockIdx.x * 256 + tid;      // float4 index, < 12*HW4
    const int plane = blockIdx.x >> 10;          // 1024 blocks per plane
    const int b = plane / 3, c = plane % 3;

    if (tid < 27) {
        int e = tid / 9, s = tid % 9;
        sC[e][s] = coef[((b * 3 + e) * 3 + c) * NSEG + s];
    }
    __syncthreads();

    v4f x4 = __builtin_nontemporal_load(&batch[gid]);
    v4f r0, r1, r2;
    #pragma unroll
    for (int j = 0; j < 4; ++j) {
        float x = x4[j];
        int s = (int)(x * 9.0f);
        s = (s < 0) ? 0 : ((s > 8) ? 8 : s);
        float xf = x - (float)s * HSTEP;
        v4f c0 = sC[0][s], c1 = sC[1][s], c2 = sC[2][s];
        r0[j] = ((c0.x * xf + c0.y) * xf + c0.z) * xf + c0.w;
        r1[j] = ((c1.x * xf + c1.y) * xf + c1.z) * xf + c1.w;
        r2[j] = ((c2.x * xf + c2.y) * xf + c2.z) * xf + c2.w;
    }

    const int pix4 = gid & (HW4 - 1);
    __builtin_nontemporal_store(r0, &out[(size_t)((0 * 4 + b) * 3 + c) * HW4 + pix4]);
    __builtin_nontemporal_store(r1, &out[(size_t)((1 * 4 + b) * 3 + c) * HW4 + pix4]);
    __builtin_nontemporal_store(r2, &out[(size_t)((2 * 4 + b) * 3 + c) * HW4 + pix4]);
}

extern "C" void kernel_launch(void* const* d_in, const int* in_sizes, int n_in,
                              void* d_out, int out_size, void* d_ws, size_t ws_size,
                              hipStream_t stream) {
    const float* batch = (const float*)d_in[0];   // [4,3,1024,1024] f32
    const float* ys    = (const float*)d_in[1];   // [4,3,3,10] f32
    float* out = (float*)d_out;
    const size_t OUT_MAIN = (size_t)3 * 4 * 3 * 1024 * 1024;  // 37,748,736
    float* splines = out + OUT_MAIN;
    v4f* coef = (v4f*)d_ws;                       // 36*9 float4 = 5184 B

    spline_precompute<<<1, 128, 0, stream>>>(ys, coef, splines);
    spline_eval<<<12288, 256, 0, stream>>>((const v4f*)batch, (const v4f*)coef, (v4f*)out);
}